// CTCLayer_71176198029896
// MI455X (gfx1250) — compile-verified
//
#include <hip/hip_runtime.h>

// CTC batch loss for B=512, T=512, C=81, L=64 (blank = C-1 = 80, PAD = 99).
// One workgroup per batch row; async global->LDS double-buffered staging of
// y_pred rows (gfx1250 ASYNCcnt path); alpha recursion in LDS in log2 domain
// (raw v_exp_f32 / v_log_f32), one barrier per time step, unroll-by-2 so the
// ping-pong alpha buffer index is compile-time constant.

#define PADTOK 99
#define NEGV   (-1e30f)
#define EPSV   (1e-7f)
#define LN2F   (0.69314718055994530942f)

constexpr int T_ = 512;
constexpr int C_ = 81;
constexpr int L_ = 64;
constexpr int S_ = 2 * L_ + 1;                    // 129
constexpr int BD = 160;                           // 5 waves (wave32)
constexpr int CHUNK_T = 64;
constexpr int NCHUNK = T_ / CHUNK_T;              // 8
constexpr int CHUNK_FLOATS = CHUNK_T * C_;        // 5184
constexpr int CHUNK_VEC4 = CHUNK_FLOATS / 4;      // 1296 (chunk base is 16B aligned)
constexpr int NITER = (CHUNK_VEC4 + BD - 1) / BD; // 9 async issues per wave per chunk

// Low 32 bits of a generic LDS pointer are the LDS byte address.
__device__ __forceinline__ unsigned lds_addr32(const void* p) {
  return (unsigned)(unsigned long long)p;
}

__device__ __forceinline__ void async_load_b128(unsigned lds_dst, const void* gsrc) {
  asm volatile("global_load_async_to_lds_b128 %0, %1, off"
               :: "v"(lds_dst), "v"(gsrc)
               : "memory");
}

__device__ __forceinline__ void wait_asynccnt_le9() {
  asm volatile("s_wait_asynccnt 9" ::: "memory");
}

// Raw hardware transcendentals: v_exp_f32 is 2^x, v_log_f32 is log2(x).
__device__ __forceinline__ float fexp2(float x) {
#if __has_builtin(__builtin_amdgcn_exp2f)
  return __builtin_amdgcn_exp2f(x);
#else
  return exp2f(x);
#endif
}
__device__ __forceinline__ float flog2(float x) {
#if __has_builtin(__builtin_amdgcn_logf)
  return __builtin_amdgcn_logf(x);
#else
  return log2f(x);
#endif
}

// One CTC forward step in log2 domain. Reads previous alphas from Ar
// (neighbors via LDS; own value carried in register a1), writes to Aw.
__device__ __forceinline__ float ctc_step(float a1, const float* __restrict__ Ar,
                                          float* __restrict__ Aw,
                                          int s, bool skip, bool valid, float lp2) {
  float a2 = Ar[s + 1];                 // alpha[s-1] (index pad -> NEG)
  float a3 = skip ? Ar[s] : NEGV;       // alpha[s-2] (index pad -> NEG)
  float m  = fmaxf(a1, fmaxf(a2, a3));  // v_max3_f32
  float sum = fexp2(a1 - m) + fexp2(a2 - m) + fexp2(a3 - m);
  float v = m + flog2(sum) + lp2;
  v = valid ? v : NEGV;
  Aw[s + 2] = v;
  return v;
}

__global__ __launch_bounds__(BD)
void ctc_loss_kernel(const int* __restrict__ y_true,
                     const float* __restrict__ y_pred,
                     float* __restrict__ out) {
  __shared__ float buf[2][CHUNK_FLOATS];   // staged y_pred rows (prob space)
  __shared__ float A[2][S_ + 35];          // alpha (log2 domain) at index s+2
  __shared__ int   lab[L_];
  __shared__ int   len_sh;

  const int b = blockIdx.x;
  const int s = threadIdx.x;

  if (s == 0) len_sh = 0;
  if (s < 2) { A[0][s] = NEGV; A[1][s] = NEGV; }
  __syncthreads();

  if (s < L_) {
    int yt = y_true[b * L_ + s];
    int good = (yt != PADTOK);
    lab[s] = good ? yt : 0;
    if (good) atomicAdd(&len_sh, 1);
  }
  __syncthreads();

  const int len = len_sh;
  const bool oddst = (s & 1) != 0;
  int l = oddst ? ((s - 1) >> 1) : 0;
  if (l > L_ - 1) l = L_ - 1;               // padding threads: clamp, masked anyway
  const int c = oddst ? lab[l] : (C_ - 1);  // emission channel for this state
  const bool valid = (s < 2 * len + 1);
  bool skip = false;
  if (oddst) skip = (l == 0) ? true : (lab[l] != lab[l - 1]);

  // ---- async staging pipeline -------------------------------------------
  const float4* gbase = (const float4*)y_pred + (size_t)b * (T_ * C_ / 4);

  // prologue: issue chunk 0 (uniform trip count; clamped index => no divergence,
  // so per-wave ASYNCcnt is exactly NITER per chunk)
  for (int i = 0; i < NITER; ++i) {
    int idx = s + i * BD; if (idx > CHUNK_VEC4 - 1) idx = CHUNK_VEC4 - 1;
    async_load_b128(lds_addr32(&buf[0][idx * 4]), gbase + idx);
  }

  float a1 = NEGV;
  float* __restrict__ A0 = A[0];
  float* __restrict__ A1 = A[1];

  for (int k = 0; k < NCHUNK; ++k) {
    // issue chunk k+1 (last chunk issues a clamped dummy so the count stays uniform)
    int knext = k + 1; if (knext > NCHUNK - 1) knext = NCHUNK - 1;
    const float4* gp = gbase + (size_t)knext * CHUNK_VEC4;
    float* dbuf = buf[(k + 1) & 1];
    for (int i = 0; i < NITER; ++i) {
      int idx = s + i * BD; if (idx > CHUNK_VEC4 - 1) idx = CHUNK_VEC4 - 1;
      async_load_b128(lds_addr32(&dbuf[idx * 4]), gp + idx);
    }
    wait_asynccnt_le9();       // <= NITER outstanding => chunk k landed (in-order)
    __syncthreads();

    const float* lpp = &buf[k & 1][c];

    if (k == 0) {
      // t = 0 init: alpha0[0] = lp(0,blank), alpha0[1] = lp(0,label0), else NEG
      float lp2 = flog2(lpp[0] + EPSV);
      float v = (s <= 1) ? lp2 : NEGV;
      if (!valid) v = NEGV;
      A0[s + 2] = v;
      a1 = v;
      __syncthreads();
      // t = 1 (odd): read A0, write A1
      a1 = ctc_step(a1, A0, A1, s, skip, valid, flog2(lpp[1 * C_] + EPSV));
      __syncthreads();
      // t = 2..63: even reads A1 writes A0, odd reads A0 writes A1
      for (int tt = 2; tt < CHUNK_T; tt += 2) {
        a1 = ctc_step(a1, A1, A0, s, skip, valid, flog2(lpp[tt * C_] + EPSV));
        __syncthreads();
        a1 = ctc_step(a1, A0, A1, s, skip, valid, flog2(lpp[(tt + 1) * C_] + EPSV));
        __syncthreads();
      }
    } else {
      // chunk starts at even t: read A1 write A0, then alternate
      for (int tt = 0; tt < CHUNK_T; tt += 2) {
        a1 = ctc_step(a1, A1, A0, s, skip, valid, flog2(lpp[tt * C_] + EPSV));
        __syncthreads();
        a1 = ctc_step(a1, A0, A1, s, skip, valid, flog2(lpp[(tt + 1) * C_] + EPSV));
        __syncthreads();
      }
    }
  }

  // t = T-1 = 511 is odd => final alphas live in A1
  if (s == 0) {
    int e1 = 2 * len;
    int e2 = 2 * len - 1; if (e2 < 0) e2 = 0;
    float x1 = A1[e1 + 2];
    float x2 = A1[e2 + 2];
    float m = fmaxf(x1, x2);
    float ll2 = m + flog2(fexp2(x1 - m) + fexp2(x2 - m));
    out[b] = -(ll2 * LN2F);   // back to natural-log units
  }
}

extern "C" void kernel_launch(void* const* d_in, const int* in_sizes, int n_in,
                              void* d_out, int out_size, void* d_ws, size_t ws_size,
                              hipStream_t stream) {
  (void)in_sizes; (void)n_in; (void)d_ws; (void)ws_size;
  const int*   y_true = (const int*)d_in[0];    // [B, L] int32
  const float* y_pred = (const float*)d_in[1];  // [B, T, C] float32 (softmax probs)
  float* out = (float*)d_out;                   // [B, 1] float32
  const int B = out_size;                       // 512
  ctc_loss_kernel<<<B, BD, 0, stream>>>(y_true, y_pred, out);
}